// LRGCNBranch_43671227466231
// MI455X (gfx1250) — compile-verified
//
#include <hip/hip_runtime.h>

// ---------------- problem constants (match reference) ----------------
#define IN_DIM   256
#define H_DIM    128
#define OUT_DIM  (3 * H_DIM)
#define LN_EPS   1e-5f

typedef __attribute__((ext_vector_type(2))) float v2f;
typedef __attribute__((ext_vector_type(8))) float v8f;

// ---------------- zero-fill scratch (h1/h2 accumulators) ----------------
__global__ __launch_bounds__(256) void zero_f32x4(float4* __restrict__ p, long n4) {
    long i = (long)blockIdx.x * blockDim.x + threadIdx.x;
    if (i < n4) p[i] = make_float4(0.f, 0.f, 0.f, 0.f);
}

// ---------------- h0 = x @ W^T via V_WMMA_F32_16X16X4_F32 ----------------
// Block: 256 threads = 8 waves. Wave w computes the 16x16 tile
// rows [blk*16, blk*16+16), cols [w*16, w*16+16). K = IN_DIM in steps of 4.
//
// A (16x4 f32, 2 VGPRs): lane l holds A[l&15][2*(l>>4)+v] in component v.
// B (4x16 f32, 2 VGPRs): lane l holds B[2*(l>>4)+v][l&15]; B[k][n] = W[col+n][k].
// D (16x16 f32, 8 VGPRs): component i -> M = i + 8*(l>>4), N = l&15.
__global__ __launch_bounds__(256) void gemm_wmma_f32(const float* __restrict__ x,
                                                     const float* __restrict__ w,
                                                     float* __restrict__ h0) {
    const int lane = threadIdx.x & 31;
    const int wave = threadIdx.x >> 5;
    const int m    = lane & 15;
    const int kh   = lane >> 4;            // 0 or 1: which K-half of the 4-wide step
    const long rowBase = (long)blockIdx.x * 16;
    const int  colBase = wave * 16;

    const float* __restrict__ arow = x + (rowBase + m) * IN_DIM + 2 * kh;
    const float* __restrict__ brow = w + (long)(colBase + m) * IN_DIM + 2 * kh;

    v8f acc = {};
#pragma unroll 8
    for (int kk = 0; kk < IN_DIM; kk += 4) {
        v2f a = *(const v2f*)(arow + kk);
        v2f b = *(const v2f*)(brow + kk);
        // (neg_a, A, neg_b, B, c_mod, C, reuse_a, reuse_b)
        acc = __builtin_amdgcn_wmma_f32_16x16x4_f32(false, a, false, b,
                                                    (short)0, acc, false, false);
    }

    float* __restrict__ orow = h0 + (rowBase + 8 * kh) * H_DIM + colBase + m;
#pragma unroll
    for (int i = 0; i < 8; ++i) orow[(long)i * H_DIM] = acc[i];
}

// ---------------- COO SpMM: out[row[e]] += val[e] * h[col[e]] ----------------
// One wave per edge; lane covers 4 contiguous floats (32 lanes * 4 = 128 = H).
// Gather of h[col] is a fully-coalesced 512B read (L2-resident, 51.2MB < 192MB L2).
__global__ __launch_bounds__(256) void spmm_edge_atomic(const int*   __restrict__ rowi,
                                                        const int*   __restrict__ coli,
                                                        const float* __restrict__ vali,
                                                        const float* __restrict__ h,
                                                        float*       __restrict__ out,
                                                        int nedges) {
    const long tid  = (long)blockIdx.x * blockDim.x + threadIdx.x;
    const int  e    = (int)(tid >> 5);
    const int  lane = (int)(tid & 31);
    if (e >= nedges) return;
    const int   r = rowi[e];
    const int   c = coli[e];
    const float v = vali[e];
    const float4 s = *(const float4*)(h + (long)c * H_DIM + lane * 4);
    float* dst = out + (long)r * H_DIM + lane * 4;
    atomicAdd(dst + 0, v * s.x);
    atomicAdd(dst + 1, v * s.y);
    atomicAdd(dst + 2, v * s.z);
    atomicAdd(dst + 3, v * s.w);
}

// ---------------- fused concat([h0,h1,h2]) + LayerNorm(384) + affine ----------------
// One wave (32 lanes) per node row; each lane owns 4 floats of each segment (12 total).
__global__ __launch_bounds__(256) void concat_layernorm(const float* __restrict__ h0,
                                                        const float* __restrict__ h1,
                                                        const float* __restrict__ h2,
                                                        const float* __restrict__ gamma,
                                                        const float* __restrict__ beta,
                                                        float* __restrict__ out,
                                                        int nrows) {
    const int lane = threadIdx.x & 31;
    const int row  = blockIdx.x * (blockDim.x >> 5) + (threadIdx.x >> 5);
    if (row >= nrows) return;

    const long base = (long)row * H_DIM + lane * 4;
    const float4 a = *(const float4*)(h0 + base);
    const float4 b = *(const float4*)(h1 + base);
    const float4 c = *(const float4*)(h2 + base);

    float s = a.x + a.y + a.z + a.w + b.x + b.y + b.z + b.w + c.x + c.y + c.z + c.w;
    float q = a.x * a.x + a.y * a.y + a.z * a.z + a.w * a.w
            + b.x * b.x + b.y * b.y + b.z * b.z + b.w * b.w
            + c.x * c.x + c.y * c.y + c.z * c.z + c.w * c.w;
#pragma unroll
    for (int off = 16; off > 0; off >>= 1) {
        s += __shfl_xor(s, off, 32);
        q += __shfl_xor(q, off, 32);
    }
    const float inv = 1.0f / (float)OUT_DIM;
    const float mean = s * inv;
    const float var  = q * inv - mean * mean;
    const float rs   = rsqrtf(var + LN_EPS);

    const int go = lane * 4;
    const float4 g0 = *(const float4*)(gamma + go);
    const float4 g1 = *(const float4*)(gamma + H_DIM + go);
    const float4 g2 = *(const float4*)(gamma + 2 * H_DIM + go);
    const float4 b0 = *(const float4*)(beta + go);
    const float4 b1 = *(const float4*)(beta + H_DIM + go);
    const float4 b2 = *(const float4*)(beta + 2 * H_DIM + go);

    float* o = out + (long)row * OUT_DIM;
    float4 r0, r1, r2;
    r0.x = (a.x - mean) * rs * g0.x + b0.x;  r0.y = (a.y - mean) * rs * g0.y + b0.y;
    r0.z = (a.z - mean) * rs * g0.z + b0.z;  r0.w = (a.w - mean) * rs * g0.w + b0.w;
    r1.x = (b.x - mean) * rs * g1.x + b1.x;  r1.y = (b.y - mean) * rs * g1.y + b1.y;
    r1.z = (b.z - mean) * rs * g1.z + b1.z;  r1.w = (b.w - mean) * rs * g1.w + b1.w;
    r2.x = (c.x - mean) * rs * g2.x + b2.x;  r2.y = (c.y - mean) * rs * g2.y + b2.y;
    r2.z = (c.z - mean) * rs * g2.z + b2.z;  r2.w = (c.w - mean) * rs * g2.w + b2.w;
    *(float4*)(o + go)             = r0;
    *(float4*)(o + H_DIM + go)     = r1;
    *(float4*)(o + 2 * H_DIM + go) = r2;
}

// ---------------- host-side launcher ----------------
extern "C" void kernel_launch(void* const* d_in, const int* in_sizes, int n_in,
                              void* d_out, int out_size, void* d_ws, size_t ws_size,
                              hipStream_t stream) {
    const float* x     = (const float*)d_in[0];
    const float* fc1_w = (const float*)d_in[1];
    const int*   row1  = (const int*)  d_in[2];
    const int*   col1  = (const int*)  d_in[3];
    const float* val1  = (const float*)d_in[4];
    const int*   row2  = (const int*)  d_in[5];
    const int*   col2  = (const int*)  d_in[6];
    const float* val2  = (const float*)d_in[7];
    const float* gamma = (const float*)d_in[8];
    const float* beta  = (const float*)d_in[9];
    float*       out   = (float*)d_out;

    const int N  = in_sizes[0] / IN_DIM;   // 100000
    const int E1 = in_sizes[2];            // 1600000
    const int E2 = in_sizes[5];

    float* h0 = (float*)d_ws;
    float* h1 = h0 + (long)N * H_DIM;
    float* h2 = h1 + (long)N * H_DIM;

    // 1) zero the two SpMM accumulators (must happen every call: atomics accumulate)
    {
        long n4 = (2L * N * H_DIM) / 4;
        int blocks = (int)((n4 + 255) / 256);
        zero_f32x4<<<blocks, 256, 0, stream>>>((float4*)h1, n4);
    }

    // 2) dense GEMM h0 = x @ W^T with fp32 WMMA (N divisible by 16: 100000/16 = 6250)
    gemm_wmma_f32<<<N / 16, 256, 0, stream>>>(x, fc1_w, h0);

    // 3) two COO SpMM hops, edge-parallel with float atomics
    {
        long t1 = (long)E1 * 32;
        long t2 = (long)E2 * 32;
        spmm_edge_atomic<<<(int)((t1 + 255) / 256), 256, 0, stream>>>(row1, col1, val1, h0, h1, E1);
        spmm_edge_atomic<<<(int)((t2 + 255) / 256), 256, 0, stream>>>(row2, col2, val2, h0, h2, E2);
    }

    // 4) fused concat + LayerNorm + affine (8 rows per 256-thread block)
    {
        int rowsPerBlock = 256 / 32;
        int blocks = (N + rowsPerBlock - 1) / rowsPerBlock;
        concat_layernorm<<<blocks, 256, 0, stream>>>(h0, h1, h2, gamma, beta, out, N);
    }
}